// MaxBprLoss_55293408968973
// MI455X (gfx1250) — compile-verified
//
#include <hip/hip_runtime.h>
#include <hip/hip_bf16.h>
#include <math.h>

typedef __attribute__((ext_vector_type(2))) float v2f;
typedef __attribute__((ext_vector_type(4))) float v4f;
typedef __attribute__((ext_vector_type(8))) float v8f;

#define BATCH 1024
#define VOCAB 100000
#define VOC4  25000      // VOCAB / 4, exact
#define DIM   64
#define K_TOP 100
#define NBINS 4096
#define CAP   4096

// ---------------------------------------------------------------------------
// Kernel 1: logits = inputs (B x D) * emb^T (D x V), f32 WMMA 16x16x4.
// Grid: (V/16, B/128), 256 threads = 8 waves; wave w owns b-rows
// [blockIdx.y*128 + w*16, +16) and the block's single v-tile of 16 columns.
// emb (25.6MB) + inputs (256KB) live in L2; the 410MB logits store dominates.
// (Round-1 codegen confirmed: 16x v_wmma fully unrolled, clause'd b64 loads.)
// ---------------------------------------------------------------------------
__global__ __launch_bounds__(256) void bpr_gemm_kernel(
    const float* __restrict__ inp, const float* __restrict__ emb,
    float* __restrict__ logits)
{
    const int lane = threadIdx.x & 31;
    const int wave = threadIdx.x >> 5;
    const int half = lane >> 4;     // 0: lanes 0-15, 1: lanes 16-31
    const int l16  = lane & 15;

    const int vBase = blockIdx.x * 16;
    const int bBase = blockIdx.y * 128 + wave * 16;

    // A frag: lane -> row M=l16, K = k0 + 2*half + {0,1}
    const float* aPtr = inp + (size_t)(bBase + l16) * DIM + 2 * half;
    // B frag: lane -> col N=l16 (= emb row vBase+l16), K = k0 + 2*half + {0,1}
    const float* bPtr = emb + (size_t)(vBase + l16) * DIM + 2 * half;

    v8f c = {};
#pragma unroll
    for (int k0 = 0; k0 < DIM; k0 += 4) {
        v2f a = *(const v2f*)(aPtr + k0);
        v2f b = *(const v2f*)(bPtr + k0);
        c = __builtin_amdgcn_wmma_f32_16x16x4_f32(
                false, a, false, b, (short)0, c, false, false);
    }

    // C layout: VGPR r holds M = r + 8*half, N = l16
    float* outP = logits + (size_t)(bBase + 8 * half) * VOCAB + vBase + l16;
#pragma unroll
    for (int r = 0; r < 8; ++r)
        outP[(size_t)r * VOCAB] = c[r];
}

// ---------------------------------------------------------------------------
// Order-preserving float -> uint key (descending float == descending key)
// ---------------------------------------------------------------------------
__device__ __forceinline__ unsigned okey(float x) {
    unsigned u = __float_as_uint(x);
    return (u & 0x80000000u) ? ~u : (u | 0x80000000u);
}

// loss term of one selected negative v: -log_sigmoid(pos - v) = softplus(v - pos)
__device__ __forceinline__ float softplus_f(float x) {
    float ax = fabsf(x);
    return fmaxf(x, 0.0f) + log1pf(expf(-ax));
}

// ---------------------------------------------------------------------------
// Kernel 2: per-row (one block per row, 256 threads) top-100 of masked logits
// via 12-bit key histogram select + boundary-bin extraction; writes per-row
// loss sum into rowLoss[row]. Both full-row scans use b128 loads (row byte
// length 400000 is 16B-multiple -> every row base stays 16B aligned) plus a
// stride-ahead global prefetch. Deterministic: integer LDS atomics only;
// float sums use fixed-order tree reductions; boundary ties are equal values.
// ---------------------------------------------------------------------------
__global__ __launch_bounds__(256) void bpr_select_loss_kernel(
    const float* __restrict__ logits, const int* __restrict__ label,
    float* __restrict__ rowLoss)
{
    __shared__ unsigned hist[NBINS];
    __shared__ float    cand[CAP];
    __shared__ float    redV[256];
    __shared__ int      redI[256];
    __shared__ unsigned s_cnt;
    __shared__ int      s_bin, s_krem;
    __shared__ unsigned s_pref;
    __shared__ float    s_pos;

    const int tid = threadIdx.x;
    const int row = blockIdx.x;
    const float* Lrow = logits + (size_t)row * VOCAB;
    const v4f*   Lrow4 = (const v4f*)Lrow;
    const int lab = label[row];

    if (tid == 0) { s_pos = Lrow[lab]; s_cnt = 0; }
    for (int i = tid; i < NBINS; i += 256) hist[i] = 0;
    __syncthreads();
    const float pos = s_pos;

    // ---- Pass A: histogram of top-12 key bits over masked row ----
    for (int q = tid; q < VOC4; q += 256) {
        __builtin_prefetch(Lrow4 + q + 1024, 0, 1);
        v4f x4 = Lrow4[q];
        int c0 = q * 4;
#pragma unroll
        for (int j = 0; j < 4; ++j) {
            float x = x4[j];
            if (c0 + j == lab) x = 0.0f;     // multiplicative one-hot mask
            atomicAdd(&hist[okey(x) >> 20], 1u);
        }
    }
    __syncthreads();
    if (tid == 0) {
        unsigned cum = 0; int b = NBINS - 1;
        for (; b > 0; --b) {
            if (cum + hist[b] >= (unsigned)K_TOP) break;
            cum += hist[b];
        }
        s_bin = b; s_krem = K_TOP - (int)cum;
    }
    __syncthreads();
    int bin = s_bin, krem = s_krem;

    // ---- Pass B: accumulate strictly-above, compact boundary bin ----
    float part = 0.0f;
    for (int q = tid; q < VOC4; q += 256) {
        __builtin_prefetch(Lrow4 + q + 1024, 0, 1);
        v4f x4 = Lrow4[q];
        int c0 = q * 4;
#pragma unroll
        for (int j = 0; j < 4; ++j) {
            float x = x4[j];
            if (c0 + j == lab) x = 0.0f;
            unsigned k12 = okey(x) >> 20;
            if ((int)k12 > bin) {
                part += softplus_f(x - pos);
            } else if ((int)k12 == bin) {
                unsigned idx = atomicAdd(&s_cnt, 1u);
                if (idx < CAP) cand[idx] = x;
            }
        }
    }
    __syncthreads();
    unsigned cnt = s_cnt;

    // ---- Fallback: refine to 24-bit prefix (exact) if boundary overflowed ----
    if (cnt > CAP) {
        __syncthreads();
        for (int i = tid; i < NBINS; i += 256) hist[i] = 0;
        if (tid == 0) s_cnt = 0;
        __syncthreads();
        for (int cidx = tid; cidx < VOCAB; cidx += 256) {
            float x = Lrow[cidx];
            if (cidx == lab) x = 0.0f;
            unsigned k = okey(x);
            if ((int)(k >> 20) == bin)
                atomicAdd(&hist[(k >> 8) & 0xFFFu], 1u);
        }
        __syncthreads();
        if (tid == 0) {
            unsigned cum = 0; int b = NBINS - 1;
            for (; b > 0; --b) {
                if (cum + hist[b] >= (unsigned)krem) break;
                cum += hist[b];
            }
            s_pref = ((unsigned)bin << 12) | (unsigned)b;
            s_krem = krem - (int)cum;
        }
        __syncthreads();
        unsigned pref = s_pref; krem = s_krem;
        part = 0.0f;
        for (int cidx = tid; cidx < VOCAB; cidx += 256) {
            float x = Lrow[cidx];
            if (cidx == lab) x = 0.0f;
            unsigned k24 = okey(x) >> 8;
            if (k24 > pref) {
                part += softplus_f(x - pos);
            } else if (k24 == pref) {
                unsigned idx = atomicAdd(&s_cnt, 1u);
                if (idx < CAP) cand[idx] = x;
            }
        }
        __syncthreads();
        cnt = s_cnt;
    }
    if (cnt > CAP) cnt = CAP;  // astronomically unlikely double overflow

    // ---- Extract krem largest boundary candidates (block argmax loop) ----
    float bsum = 0.0f;  // only meaningful on tid==0
    for (int it = 0; it < krem; ++it) {
        float lv = -INFINITY; int li = -1;
        for (int i = tid; i < (int)cnt; i += 256) {
            float v = cand[i];
            if (v > lv) { lv = v; li = i; }
        }
        redV[tid] = lv; redI[tid] = li;
        __syncthreads();
        for (int s = 128; s > 0; s >>= 1) {
            if (tid < s) {
                float ov = redV[tid + s]; int oi = redI[tid + s];
                if (oi >= 0 && (redI[tid] < 0 || ov > redV[tid] ||
                                (ov == redV[tid] && oi < redI[tid]))) {
                    redV[tid] = ov; redI[tid] = oi;
                }
            }
            __syncthreads();
        }
        if (tid == 0 && redI[0] >= 0) {
            bsum += softplus_f(redV[0] - pos);
            cand[redI[0]] = -INFINITY;
        }
        __syncthreads();
    }

    // ---- Deterministic tree-reduce of the strictly-above partials ----
    redV[tid] = part;
    __syncthreads();
    for (int s = 128; s > 0; s >>= 1) {
        if (tid < s) redV[tid] += redV[tid + s];
        __syncthreads();
    }
    if (tid == 0) rowLoss[row] = redV[0] + bsum;
}

// ---------------------------------------------------------------------------
// Kernel 3: mean over the 1024 per-row losses -> d_out[B*V]
// ---------------------------------------------------------------------------
__global__ __launch_bounds__(256) void bpr_finalize_kernel(
    const float* __restrict__ rowLoss, float* __restrict__ lossOut)
{
    __shared__ float red[256];
    const int tid = threadIdx.x;
    float s = 0.0f;
    for (int i = tid; i < BATCH; i += 256) s += rowLoss[i];
    red[tid] = s;
    __syncthreads();
    for (int st = 128; st > 0; st >>= 1) {
        if (tid < st) red[tid] += red[tid + st];
        __syncthreads();
    }
    if (tid == 0) lossOut[0] = red[0] / (float)BATCH;
}

// ---------------------------------------------------------------------------
extern "C" void kernel_launch(void* const* d_in, const int* in_sizes, int n_in,
                              void* d_out, int out_size, void* d_ws, size_t ws_size,
                              hipStream_t stream) {
    (void)in_sizes; (void)n_in; (void)out_size; (void)ws_size;
    const float* inp   = (const float*)d_in[0];   // (1024, 64)
    const float* emb   = (const float*)d_in[1];   // (100000, 64)
    const int*   label = (const int*)d_in[2];     // (1024,)
    float* logits  = (float*)d_out;                         // first B*V floats
    float* lossOut = (float*)d_out + (size_t)BATCH * VOCAB; // then scalar loss
    float* rowLoss = (float*)d_ws;                          // 1024 floats scratch

    dim3 gGrid(VOCAB / 16, BATCH / 128);   // 6250 x 8
    bpr_gemm_kernel<<<gGrid, 256, 0, stream>>>(inp, emb, logits);
    bpr_select_loss_kernel<<<BATCH, 256, 0, stream>>>(logits, label, rowLoss);
    bpr_finalize_kernel<<<1, 256, 0, stream>>>(rowLoss, lossOut);
}